// SimpleAttention_78048145703024
// MI455X (gfx1250) — compile-verified
//
#include <hip/hip_runtime.h>
#include <math.h>
#include <stdint.h>

// Problem constants (match reference)
#define BB   16
#define SS   4096
#define HH   1024
#define KB   8            // K-block staged in LDS per pipeline stage
#define PADA 12           // padded LDS stride: 16B-aligned b128 stores, conflict-free frags
#define MT   64           // rows per workgroup (4 m-subtiles)
#define NT   512          // cols per workgroup (half of H)
#define NBLK (HH / KB)    // 128 K-blocks

typedef __attribute__((ext_vector_type(2))) float v2f;
typedef __attribute__((ext_vector_type(8))) float v8f;
typedef __attribute__((vector_size(16))) int vi4;   // async b128 payload type
typedef __attribute__((vector_size(8)))  int vi2;   // async b64 payload type

// ---- CDNA5 feature probes (guarded so either toolchain compiles) ----------
#if __has_builtin(__builtin_amdgcn_global_load_async_to_lds_b128) && \
    __has_builtin(__builtin_amdgcn_global_load_async_to_lds_b64)
#define HAVE_ASYNC 1
#else
#define HAVE_ASYNC 0
#endif

#if __has_builtin(__builtin_amdgcn_s_wait_asynccnt)
#define WAIT_ASYNC(n) __builtin_amdgcn_s_wait_asynccnt((short)(n))
#else
#define WAIT_ASYNC(n) asm volatile("s_wait_asynccnt %0" ::"n"(n) : "memory")
#endif

#if __has_builtin(__builtin_amdgcn_tanhf)
#define TANHF(v) __builtin_amdgcn_tanhf(v)
#elif __has_builtin(__builtin_amdgcn_tanh_f32)
#define TANHF(v) __builtin_amdgcn_tanh_f32(v)
#else
#define TANHF(v) tanhf(v)
#endif

#define AS1 __attribute__((address_space(1)))
#define AS3 __attribute__((address_space(3)))

__device__ __forceinline__ static void cp_async_b128(void* lds, const void* g) {
#if HAVE_ASYNC
  __builtin_amdgcn_global_load_async_to_lds_b128(
      (AS1 vi4*)(uintptr_t)g, (AS3 vi4*)(uintptr_t)lds, 0, 0);
#else
  (void)lds; (void)g;
#endif
}
__device__ __forceinline__ static void cp_async_b64(void* lds, const void* g) {
#if HAVE_ASYNC
  __builtin_amdgcn_global_load_async_to_lds_b64(
      (AS1 vi2*)(uintptr_t)g, (AS3 vi2*)(uintptr_t)lds, 0, 0);
#else
  (void)lds; (void)g;
#endif
}

// ---------------------------------------------------------------------------
// Phase 1: fused GEMM (x @ W^T) + bias + tanh + dot(key_vec)
//          -> partial scores per column-half.
// One WG: 64 rows x 512 cols; K pipelined in blocks of 8 through double-
// buffered LDS filled by async global->LDS copies (ASYNCcnt tracked).
// ---------------------------------------------------------------------------
__global__ __launch_bounds__(256) void attn_scores_kernel(
    const float* __restrict__ x, const int* __restrict__ lengths,
    const float* __restrict__ W, const float* __restrict__ bias,
    const float* __restrict__ keyv, float* __restrict__ scoresPart)
{
  __shared__ float Asm[2][MT * PADA];   // 2 x 64x8  x-tile   (6 KB)
  __shared__ float Wsm[2][NT * PADA];   // 2 x 512x8 W-tile   (48 KB)
  __shared__ float wsum[8 * MT];        // per-wave row partials (2 KB)

  const int tid     = threadIdx.x;
  const int rowTile = blockIdx.x >> 1;
  const int half    = blockIdx.x & 1;      // which 512-column half of W
  const int rowBase = rowTile * MT;        // tile lies entirely within one b
  const int nbase   = half * NT;
  const int bidx    = rowBase / SS;
  const int srow    = rowBase % SS;
  if (srow >= lengths[bidx]) return;       // fully-masked tile: skip all work

  const int w    = tid >> 5;               // wave id 0..7
  const int lane = tid & 31;
  const int m    = lane & 15;
  const int hi   = lane >> 4;
  const int wcol = w << 6;                 // this wave's 64-column slab

  v8f c[4][4];
  #pragma unroll
  for (int ms = 0; ms < 4; ++ms)
    #pragma unroll
    for (int j = 0; j < 4; ++j) c[ms][j] = {};

  // per-thread staging coordinates (uniform: every thread issues 4+1 copies)
  const int aRow = tid >> 2;               // 0..63
  const int aKk  = (tid & 3) << 1;         // 0,2,4,6

  auto issue = [&](int buf, int k0) {
    #pragma unroll
    for (int rep = 0; rep < 4; ++rep) {    // W tile: 1024 float4 / 256 thr
      const int e   = (rep << 8) + tid;
      const int n   = e >> 1;              // 0..511
      const int kk4 = (e & 1) << 2;        // 0 or 4
      cp_async_b128(&Wsm[buf][n * PADA + kk4],
                    &W[(size_t)(nbase + n) * HH + k0 + kk4]);
    }
    cp_async_b64(&Asm[buf][aRow * PADA + aKk],
                 &x[(size_t)(rowBase + aRow) * HH + k0 + aKk]);
  };

  auto compute = [&](int cur) {
    #pragma unroll
    for (int ks = 0; ks < KB / 4; ++ks) {
      const int ka = (ks << 2) + (hi << 1);
      v2f a[4];
      #pragma unroll
      for (int ms = 0; ms < 4; ++ms) {
        const float2 t = *(const float2*)&Asm[cur][((ms << 4) + m) * PADA + ka];
        a[ms].x = t.x; a[ms].y = t.y;
      }
      #pragma unroll
      for (int j = 0; j < 4; ++j) {
        const int nl = wcol + (j << 4) + m;
        const float2 tb = *(const float2*)&Wsm[cur][nl * PADA + ka];
        v2f bf; bf.x = tb.x; bf.y = tb.y;
        #pragma unroll
        for (int ms = 0; ms < 4; ++ms)
          c[ms][j] = __builtin_amdgcn_wmma_f32_16x16x4_f32(
              false, a[ms], false, bf, (short)0, c[ms][j], false, false);
      }
    }
  };

#if HAVE_ASYNC
  issue(0, 0);                              // prologue: fill buffer 0
  for (int blk = 0; blk < NBLK; ++blk) {
    const int cur = blk & 1;
    if (blk + 1 < NBLK) {
      issue(cur ^ 1, (blk + 1) * KB);       // overlap next stage with compute
      WAIT_ASYNC(5);                        // cur's 5 copies done (in-order)
    } else {
      WAIT_ASYNC(0);
    }
    __syncthreads();                        // all waves' cur data visible
    compute(cur);
    __syncthreads();                        // done reading before overwrite
  }
#else
  // synchronous fallback: single-buffer staging
  for (int blk = 0; blk < NBLK; ++blk) {
    const int k0 = blk * KB;
    __syncthreads();
    #pragma unroll
    for (int rep = 0; rep < 4; ++rep) {
      const int e   = (rep << 8) + tid;
      const int n   = e >> 1;
      const int kk4 = (e & 1) << 2;
      const float4 v = *(const float4*)(&W[(size_t)(nbase + n) * HH + k0 + kk4]);
      float* dst = &Wsm[0][n * PADA + kk4];
      *(float2*)(dst)     = make_float2(v.x, v.y);
      *(float2*)(dst + 2) = make_float2(v.z, v.w);
    }
    {
      const float2 v = *(const float2*)(&x[(size_t)(rowBase + aRow) * HH + k0 + aKk]);
      *(float2*)&Asm[0][aRow * PADA + aKk] = v;
    }
    __syncthreads();
    compute(0);
  }
#endif

  // ---- epilogue: score_row += tanh(acc + bias[n]) * key[n] ----
  float pr[4][8];
  #pragma unroll
  for (int ms = 0; ms < 4; ++ms)
    #pragma unroll
    for (int r = 0; r < 8; ++r) pr[ms][r] = 0.0f;

  #pragma unroll
  for (int j = 0; j < 4; ++j) {
    const int   n  = nbase + wcol + (j << 4) + m;
    const float bb = bias[n];
    const float kv = keyv[n];
    #pragma unroll
    for (int ms = 0; ms < 4; ++ms)
      #pragma unroll
      for (int r = 0; r < 8; ++r)
        pr[ms][r] += TANHF(c[ms][j][r] + bb) * kv;
  }

  // deterministic cross-lane reduce within each 16-lane half
  #pragma unroll
  for (int ms = 0; ms < 4; ++ms)
    #pragma unroll
    for (int r = 0; r < 8; ++r) {
      float v = pr[ms][r];
      v += __shfl_xor(v, 1);
      v += __shfl_xor(v, 2);
      v += __shfl_xor(v, 4);
      v += __shfl_xor(v, 8);
      pr[ms][r] = v;
    }
  if (m == 0) {                 // lanes 0 and 16 hold half-wave totals
    #pragma unroll
    for (int ms = 0; ms < 4; ++ms)
      #pragma unroll
      for (int r = 0; r < 8; ++r)
        wsum[w * MT + (ms << 4) + (hi << 3) + r] = pr[ms][r];
  }
  __syncthreads();
  if (tid < MT) {               // fixed-order combine across waves
    float sc = 0.0f;
    #pragma unroll
    for (int ww = 0; ww < 8; ++ww) sc += wsum[ww * MT + tid];
    scoresPart[(size_t)half * (BB * SS) + rowBase + tid] = sc;
  }
}

// ---------------------------------------------------------------------------
// Phase 2: combine column-half partials, masked softmax; attn -> p0 in place
// ---------------------------------------------------------------------------
__global__ __launch_bounds__(256) void attn_softmax_kernel(
    const int* __restrict__ lengths, float* __restrict__ p0,
    const float* __restrict__ p1)
{
  __shared__ float red[256];
  const int b   = blockIdx.x;
  const int tid = threadIdx.x;
  const int len = lengths[b];
  float*       s0 = p0 + (size_t)b * SS;
  const float* s1 = p1 + (size_t)b * SS;

  float mx = -INFINITY;
  for (int i = tid; i < len; i += 256) mx = fmaxf(mx, s0[i] + s1[i]);
  red[tid] = mx; __syncthreads();
  #pragma unroll
  for (int st = 128; st > 0; st >>= 1) {
    if (tid < st) red[tid] = fmaxf(red[tid], red[tid + st]);
    __syncthreads();
  }
  mx = red[0]; __syncthreads();

  float sum = 0.0f;
  for (int i = tid; i < len; i += 256) sum += __expf(s0[i] + s1[i] - mx);
  red[tid] = sum; __syncthreads();
  #pragma unroll
  for (int st = 128; st > 0; st >>= 1) {
    if (tid < st) red[tid] += red[tid + st];
    __syncthreads();
  }
  sum = red[0]; __syncthreads();

  const float inv = 1.0f / sum;
  for (int i = tid; i < len; i += 256) s0[i] = __expf(s0[i] + s1[i] - mx) * inv;
}

// ---------------------------------------------------------------------------
// Phase 3: out[b,h] = sum_{s<len} attn[b,s] * x[b,s,h]  (memory-bound)
// ---------------------------------------------------------------------------
__global__ __launch_bounds__(256) void attn_wsum_kernel(
    const float* __restrict__ x, const int* __restrict__ lengths,
    const float* __restrict__ attn, float* __restrict__ out)
{
  const int b   = blockIdx.x >> 2;
  const int h   = ((blockIdx.x & 3) << 8) + threadIdx.x;
  const int len = lengths[b];
  const float* ab = attn + (size_t)b * SS;
  const float* xb = x + ((size_t)b * SS) * HH + h;

  float acc = 0.0f;
  int s = 0;
  for (; s + 4 <= len; s += 4) {
    acc += ab[s]     * xb[(size_t)s * HH];
    acc += ab[s + 1] * xb[(size_t)(s + 1) * HH];
    acc += ab[s + 2] * xb[(size_t)(s + 2) * HH];
    acc += ab[s + 3] * xb[(size_t)(s + 3) * HH];
  }
  for (; s < len; ++s) acc += ab[s] * xb[(size_t)s * HH];
  out[b * HH + h] = acc;
}

// ---------------------------------------------------------------------------
extern "C" void kernel_launch(void* const* d_in, const int* in_sizes, int n_in,
                              void* d_out, int out_size, void* d_ws, size_t ws_size,
                              hipStream_t stream) {
  const float* x    = (const float*)d_in[0];  // [B,S,H]
  const int*   len  = (const int*)  d_in[1];  // [B]
  const float* W    = (const float*)d_in[2];  // [H,H]
  const float* bias = (const float*)d_in[3];  // [H]
  const float* keyv = (const float*)d_in[4];  // [H]
  float* out = (float*)d_out;                 // [B,H]
  float* p0  = (float*)d_ws;                  // partial scores half 0 / attn
  float* p1  = p0 + (size_t)BB * SS;          // partial scores half 1

  attn_scores_kernel<<<(BB * SS / MT) * 2, 256, 0, stream>>>(x, len, W, bias, keyv, p0);
  attn_softmax_kernel<<<BB, 256, 0, stream>>>(len, p0, p1);
  attn_wsum_kernel<<<BB * (HH / 256), 256, 0, stream>>>(x, len, p0, out);

  (void)in_sizes; (void)n_in; (void)out_size; (void)ws_size;
}